// InvertSingleDirection_2087354106275
// MI455X (gfx1250) — compile-verified
//
#include <hip/hip_runtime.h>
#include <hip/hip_bf16.h>
#include <cstdint>

// Shapes fixed by the reference: mag_field (8,128,128,128,1) f32, direction (8,1,3) f32.
static constexpr int HH = 128, WW = 128, DD = 128;
static constexpr int VOX = HH * WW * DD;

__device__ __forceinline__ float sgpr_bcast(float x) {
  return __int_as_float(__builtin_amdgcn_readfirstlane(__float_as_int(x)));
}

// Trilinear sample with fill_value=0 semantics, branchless:
// clamp indices for the (always-safe) loads, zero the per-axis weight of any
// out-of-range corner -> product weight is 0 exactly like the reference's
// clip + where(valid, gathered, 0) * w.
__device__ __forceinline__ float tri_sample_fill0(const float* __restrict__ vol,
                                                  float x, float y, float z) {
  const float fx = floorf(x), fy = floorf(y), fz = floorf(z);
  const int ix = (int)fx, iy = (int)fy, iz = (int)fz;
  const float ax = x - fx, ay = y - fy, az = z - fz;

  const int ix0 = min(max(ix,     0), HH - 1), ix1 = min(max(ix + 1, 0), HH - 1);
  const int iy0 = min(max(iy,     0), WW - 1), iy1 = min(max(iy + 1, 0), WW - 1);
  const int iz0 = min(max(iz,     0), DD - 1), iz1 = min(max(iz + 1, 0), DD - 1);

  const float wx0 = ((unsigned)ix       < (unsigned)HH) ? 1.0f - ax : 0.0f;
  const float wx1 = ((unsigned)(ix + 1) < (unsigned)HH) ? ax        : 0.0f;
  const float wy0 = ((unsigned)iy       < (unsigned)WW) ? 1.0f - ay : 0.0f;
  const float wy1 = ((unsigned)(iy + 1) < (unsigned)WW) ? ay        : 0.0f;
  const float wz0 = ((unsigned)iz       < (unsigned)DD) ? 1.0f - az : 0.0f;
  const float wz1 = ((unsigned)(iz + 1) < (unsigned)DD) ? az        : 0.0f;

  const float* r00 = vol + (ix0 * WW + iy0) * DD;
  const float* r01 = vol + (ix0 * WW + iy1) * DD;
  const float* r10 = vol + (ix1 * WW + iy0) * DD;
  const float* r11 = vol + (ix1 * WW + iy1) * DD;

  const float s00 = wz0 * r00[iz0] + wz1 * r00[iz1];
  const float s01 = wz0 * r01[iz0] + wz1 * r01[iz1];
  const float s10 = wz0 * r10[iz0] + wz1 * r10[iz1];
  const float s11 = wz0 * r11[iz0] + wz1 * r11[iz1];

  return wx0 * (wy0 * s00 + wy1 * s01) + wx1 * (wy0 * s10 + wy1 * s11);
}

// One thread handles 4 consecutive d-voxels: one float4 mag load, 32 gathers
// (whole 64 MiB input is L2-resident on the 192 MB L2), 48 contiguous output
// bytes staged in LDS and pushed via the CDNA5 async LDS->global store path
// (ASYNCcnt) with non-temporal hint so the 192 MiB write-once output stream
// does not evict the gather working set from L2.
__global__ __launch_bounds__(256) void InvertSingleDirection_kernel(
    const float* __restrict__ mag, const float* __restrict__ dir,
    float* __restrict__ out) {
  const int b = blockIdx.y;
  const int t = blockIdx.x * 256 + threadIdx.x;     // 0 .. 128*128*32-1
  const int d4 = (t & 31) << 2;                     // d base, multiple of 4
  const int w  = (t >> 5) & (WW - 1);
  const int h  = t >> 12;

  // Per-batch direction: uniform address -> keep in SGPRs.
  const float* dp = dir + 3 * (size_t)b;
  const float dx = sgpr_bcast(dp[0]);
  const float dy = sgpr_bcast(dp[1]);
  const float dz = sgpr_bcast(dp[2]);

  const float* vol = mag + (size_t)b * VOX;
  const int vbase = (h * WW + w) * DD + d4;         // 16B aligned
  const float4 m4 = *reinterpret_cast<const float4*>(vol + vbase);
  const float mm[4] = {m4.x, m4.y, m4.z, m4.w};

  float o[12];
#pragma unroll
  for (int j = 0; j < 4; ++j) {
    const float m = mm[j];
    const float warped = tri_sample_fill0(vol,
                                          (float)h        + m * dx,
                                          (float)w        + m * dy,
                                          (float)(d4 + j) + m * dz);
    o[3 * j + 0] = -warped * dx;
    o[3 * j + 1] = -warped * dy;
    o[3 * j + 2] = -warped * dz;
  }

  // Stage 48B/thread in LDS, then async-store straight from LDS to HBM.
  __shared__ float4 stage[256][3];
  float4* sp = &stage[threadIdx.x][0];
  sp[0] = make_float4(o[0], o[1], o[2],  o[3]);
  sp[1] = make_float4(o[4], o[5], o[6],  o[7]);
  sp[2] = make_float4(o[8], o[9], o[10], o[11]);

  const uint32_t lds = (uint32_t)(uintptr_t)sp;     // low 32b = LDS offset
  const unsigned long long ga =
      (unsigned long long)(uintptr_t)(out + 3ull * ((size_t)b * VOX + (size_t)vbase));

  asm volatile(
      "s_wait_dscnt 0\n\t"
      "global_store_async_from_lds_b128 %0, %1, off th:TH_STORE_NT\n\t"
      "global_store_async_from_lds_b128 %0, %1, off offset:16 th:TH_STORE_NT\n\t"
      "global_store_async_from_lds_b128 %0, %1, off offset:32 th:TH_STORE_NT\n\t"
      "s_wait_asynccnt 0"
      :: "v"(ga), "v"(lds)
      : "memory");
}

extern "C" void kernel_launch(void* const* d_in, const int* in_sizes, int n_in,
                              void* d_out, int out_size, void* d_ws, size_t ws_size,
                              hipStream_t stream) {
  (void)n_in; (void)out_size; (void)d_ws; (void)ws_size;
  const float* mag = (const float*)d_in[0];   // (B,128,128,128,1)
  const float* dir = (const float*)d_in[1];   // (B,1,3)
  float* out = (float*)d_out;                 // (B,128,128,128,3)
  const int B = in_sizes[1] / 3;              // 8

  dim3 grid((HH * WW * (DD / 4)) / 256, B);   // (2048, 8)
  InvertSingleDirection_kernel<<<grid, 256, 0, stream>>>(mag, dir, out);
}